// MatrixExpEmbedding_11965778887193
// MI455X (gfx1250) — compile-verified
//
#include <hip/hip_runtime.h>
#include <math.h>

// out[n] = expm((x[n]/pi) * A) @ v  for fixed 2x2 A and 2-vec v.
// Closed form (all A/v-derived terms wave-uniform -> scalar-float pipe):
//   out = alpha(s)*v + beta(s)*w,  w = A@v,  mu = s*h,  d = |s|*sqrt(|q|)
//   alpha = e^mu*(c - sc*mu), beta = e^mu*sc*s
//   (c,sc) = (cosh d, sinh d/d) if q>=0 else (cos d, sin d/d)   [uniform branch]
// Bandwidth-bound: 16 MB in + 32 MB out => ~2.1 us @ 23.3 TB/s.
// Input is staged through LDS with CDNA5 double-buffered async copies:
//   global_load_async_to_lds_b128 (GVS) + s_wait_asynccnt + ds_load_b128.

#define INV_PI_F 0.31830988618379067154f
#define TPB 256

__device__ __forceinline__ void async_ld_b128(uint32_t lds_addr, uint32_t goff,
                                              const float* gbase) {
    // VDST = LDS byte address (per-lane), VADDR = 32-bit byte offset, SADDR = base.
    asm volatile("global_load_async_to_lds_b128 %0, %1, %2"
                 :
                 : "v"(lds_addr), "v"(goff), "s"(gbase)
                 : "memory");
}

__global__ __launch_bounds__(TPB) void matexp2x2_stream_kernel(
    const float* __restrict__ x,
    const float* __restrict__ A,   // 4 floats, row-major 2x2
    const float* __restrict__ v,   // 2 floats
    float* __restrict__ out,       // [N,2] row-major
    int n)                         // total elements N
{
    __shared__ float4 buf[2][TPB];

    // ---- wave-uniform preamble (s_load + scalar-float ops) ----
    const float a00 = A[0], a01 = A[1], a10 = A[2], a11 = A[3];
    const float v0  = v[0], v1  = v[1];
    const float h   = 0.5f * (a00 + a11);          // tr/2
    const float det = a00 * a11 - a01 * a10;
    const float q   = h * h - det;                 // d^2 = s^2 * q
    const float w0  = a00 * v0 + a01 * v1;         // w = A @ v
    const float w1  = a10 * v0 + a11 * v1;
    const bool  hyper = (q >= 0.0f);               // uniform branch select
    const float r   = sqrtf(fabsf(q));

    const int n4     = n >> 2;                     // float4 groups
    const int stride = (int)(gridDim.x * blockDim.x);
    const int tid    = (int)threadIdx.x;
    int i = (int)(blockIdx.x * blockDim.x) + tid;

    // LDS byte addresses of this lane's two staging slots
    // (generic shared pointer low 32 bits == LDS offset on gfx1250).
    const uint32_t lds_a[2] = {
        (uint32_t)(uintptr_t)&buf[0][tid],
        (uint32_t)(uintptr_t)&buf[1][tid]
    };

    if (i < n4) {
        // Prologue: stage first tile.
        async_ld_b128(lds_a[0], (uint32_t)i * 16u, x);
        int p = 0;
        for (; i < n4; i += stride) {
            // Stage next tile into the other buffer (clamp keeps it in-range
            // on the final iteration so the wait level stays constant).
            const int inext = i + stride;
            const int ipre  = (inext < n4) ? inext : i;
            async_ld_b128(lds_a[p ^ 1], (uint32_t)ipre * 16u, x);

            // Wait for the older async copy (1 newer op still outstanding).
            asm volatile("s_wait_asynccnt 0x1" ::: "memory");

            const float4 xs = buf[p][tid];         // ds_load_b128 (own slot)
            const float sx[4] = {xs.x, xs.y, xs.z, xs.w};
            float o[8];

#pragma unroll
            for (int k = 0; k < 4; ++k) {
                const float s  = sx[k] * INV_PI_F;
                const float mu = s * h;
                const float d  = fabsf(s) * r;     // |delta| or omega
                float Ac, Bc;                      // e^mu*c , e^mu*sc
                if (hyper) {
                    const float ep = __expf(mu + d);
                    const float em = __expf(mu - d);
                    Ac = 0.5f * (ep + em);
                    const float Bs = (ep - em) / (2.0f * d);
                    const float Bl = __expf(mu) * (1.0f + d * d * (1.0f / 6.0f));
                    Bc = (d > 1e-4f) ? Bs : Bl;    // sinh(d)/d, small-d safe
                } else {
                    const float emu = __expf(mu);
                    float sn, cs;
                    __sincosf(d, &sn, &cs);
                    Ac = emu * cs;
                    const float Bs = emu * sn / d;
                    const float Bl = emu * (1.0f - d * d * (1.0f / 6.0f));
                    Bc = (d > 1e-4f) ? Bs : Bl;    // sin(d)/d, small-d safe
                }
                const float alpha = Ac - Bc * mu;
                const float beta  = Bc * s;
                o[2 * k + 0] = alpha * v0 + beta * w0;
                o[2 * k + 1] = alpha * v1 + beta * w1;
            }

            float4* op = reinterpret_cast<float4*>(out + (size_t)i * 8);
            op[0] = make_float4(o[0], o[1], o[2], o[3]);   // global_store_b128
            op[1] = make_float4(o[4], o[5], o[6], o[7]);   // global_store_b128
            p ^= 1;
        }
        // Drain the dangling prefetch (s_endpgm would also wait-idle).
        asm volatile("s_wait_asynccnt 0x0" ::: "memory");
    }

    // Tail (N % 4 != 0): scalar direct path, negligible.
    for (int idx = (n4 << 2) + (int)(blockIdx.x * blockDim.x) + tid;
         idx < n; idx += stride) {
        const float s  = x[idx] * INV_PI_F;
        const float mu = s * h;
        const float d  = fabsf(s) * r;
        float Ac, Bc;
        if (hyper) {
            const float ep = __expf(mu + d);
            const float em = __expf(mu - d);
            Ac = 0.5f * (ep + em);
            const float Bs = (ep - em) / (2.0f * d);
            const float Bl = __expf(mu) * (1.0f + d * d * (1.0f / 6.0f));
            Bc = (d > 1e-4f) ? Bs : Bl;
        } else {
            const float emu = __expf(mu);
            float sn, cs;
            __sincosf(d, &sn, &cs);
            Ac = emu * cs;
            const float Bs = emu * sn / d;
            const float Bl = emu * (1.0f - d * d * (1.0f / 6.0f));
            Bc = (d > 1e-4f) ? Bs : Bl;
        }
        const float alpha = Ac - Bc * mu;
        const float beta  = Bc * s;
        out[(size_t)idx * 2 + 0] = alpha * v0 + beta * w0;
        out[(size_t)idx * 2 + 1] = alpha * v1 + beta * w1;
    }
}

extern "C" void kernel_launch(void* const* d_in, const int* in_sizes, int n_in,
                              void* d_out, int out_size, void* d_ws, size_t ws_size,
                              hipStream_t stream) {
    (void)n_in; (void)out_size; (void)d_ws; (void)ws_size;
    const float* x = (const float*)d_in[0];   // [N]
    const float* A = (const float*)d_in[1];   // [2,2]
    const float* v = (const float*)d_in[2];   // [2,1]
    float* out     = (float*)d_out;           // [N,2]
    const int n    = in_sizes[0];

    // Grid-stride: 1024 blocks x 256 threads (8 wave32/block) -> 262,144 lanes,
    // 4 float4-iterations each at N=4M; async staging covers the next tile.
    const int threads = TPB;
    int blocks = 1024;
    const int n4 = n >> 2;
    if (blocks * threads > n4 && n4 > 0) blocks = (n4 + threads - 1) / threads;
    if (blocks < 1) blocks = 1;

    matexp2x2_stream_kernel<<<blocks, threads, 0, stream>>>(x, A, v, out, n);
}